// MoELayer_81114752352735
// MI455X (gfx1250) — compile-verified
//
#include <hip/hip_runtime.h>

typedef __attribute__((ext_vector_type(2))) float v2f;
typedef __attribute__((ext_vector_type(4))) float v4f;
typedef __attribute__((ext_vector_type(8))) float v8f;
typedef __attribute__((ext_vector_type(4))) int   i32x4;

#define TOKENS 4096
#define DDIM   1024
#define FDIM   1024
#define NEXP   8

// ---- CDNA5 async global->LDS staging (guarded; sync fallback) -------------
#if defined(__has_builtin)
#if __has_builtin(__builtin_amdgcn_global_load_async_to_lds_b128)
#define ASYNC_OK 1
#endif
#endif
#ifndef ASYNC_OK
#define ASYNC_OK 0
#endif

__device__ __forceinline__ void cp16(float* dst_lds, const float* src_glob)
{
#if ASYNC_OK
    __builtin_amdgcn_global_load_async_to_lds_b128(
        (__attribute__((address_space(1))) i32x4*)(const_cast<float*>(src_glob)),
        (__attribute__((address_space(3))) i32x4*)dst_lds,
        /*imm offset=*/0, /*cpol=*/0);
#else
    *(v4f*)dst_lds = *(const v4f*)src_glob;
#endif
}

__device__ __forceinline__ void wait_async_le4()
{
#if ASYNC_OK
#if __has_builtin(__builtin_amdgcn_s_wait_asynccnt)
    __builtin_amdgcn_s_wait_asynccnt(4);
#else
    asm volatile("s_wait_asynccnt 4" ::: "memory");
#endif
#endif
}

__device__ __forceinline__ void wait_async_le0()
{
#if ASYNC_OK
#if __has_builtin(__builtin_amdgcn_s_wait_asynccnt)
    __builtin_amdgcn_s_wait_asynccnt(0);
#else
    asm volatile("s_wait_asynccnt 0" ::: "memory");
#endif
#endif
}

// ---------------------------------------------------------------------------
// Kernel 1: noisy router. One wave32 per token. Computes top-2 combine weights
// directly from logits (softmax denominator cancels in top-k renorm).
// Writes combine[T][8] (zero for non-selected experts) into workspace.
// ---------------------------------------------------------------------------
__global__ __launch_bounds__(256) void moe_router(
    const float* __restrict__ x, const float* __restrict__ nx,
    const float* __restrict__ rw, const float* __restrict__ nrw,
    float* __restrict__ combine)
{
    const int token = (blockIdx.x * blockDim.x + threadIdx.x) >> 5;
    const int lane  = threadIdx.x & 31;
    if (token >= TOKENS) return;

    const float* xr = x  + (size_t)token * DDIM;
    const float* nr = nx + (size_t)token * DDIM;

    float acc[NEXP];
#pragma unroll
    for (int e = 0; e < NEXP; ++e) acc[e] = 0.0f;

    for (int i = 0; i < DDIM / 32; ++i) {
        const int d = lane + 32 * i;
        const float xv = xr[d];
        const float nv = nr[d];
        const float* r0 = rw  + d * NEXP;
        const float* n0 = nrw + d * NEXP;
#pragma unroll
        for (int e = 0; e < NEXP; ++e)
            acc[e] += xv * r0[e] + nv * n0[e];
    }

#pragma unroll
    for (int e = 0; e < NEXP; ++e) {
        float v = acc[e];
        for (int off = 16; off > 0; off >>= 1)
            v += __shfl_xor(v, off, 32);
        acc[e] = v;
    }

    if (lane == 0) {
        int i1 = 0; float l1 = acc[0];
        for (int e = 1; e < NEXP; ++e)
            if (acc[e] > l1) { l1 = acc[e]; i1 = e; }
        int i2 = -1; float l2 = -3.402823466e38f;
        for (int e = 0; e < NEXP; ++e)
            if (e != i1 && acc[e] > l2) { l2 = acc[e]; i2 = e; }
        const float g2 = __expf(l2 - l1);        // g1 = 1
        const float inv = 1.0f / (1.0f + g2);
        float* c = combine + (size_t)token * NEXP;
#pragma unroll
        for (int e = 0; e < NEXP; ++e) c[e] = 0.0f;
        c[i1] = inv;
        c[i2] = g2 * inv;
    }
}

// ---------------------------------------------------------------------------
// Kernel 2: gated grouped GEMM, fp32 WMMA, 2x2 register blocking per wave,
// double-buffered LDS with 1-tile-ahead async pipeline:
//   iter p: issue async loads for tile p+1 -> other buffer,
//           s_wait_asynccnt <= 4 (in-order completion => tile p has landed),
//           barrier, WMMA on tile p (copies for p+1 run concurrently), barrier.
// ---------------------------------------------------------------------------
__global__ __launch_bounds__(256) void moe_expert(
    const float* __restrict__ x, const float* __restrict__ ew,
    const float* __restrict__ eb, const float* __restrict__ combine,
    float* __restrict__ out)
{
    constexpr int BM = 64, BN = 128, KC = 32;
    constexpr int XP = 36;   // row stride: banks 4*(9l%16) / +2 -> conflict-free, 16B aligned
    constexpr int WP = 136;  // 2*136 % 64 == 16 -> half-waves hit disjoint banks

    __shared__ __align__(16) float sX[2][BM][XP];
    __shared__ __align__(16) float sW[2][KC][WP];
    __shared__ float sComb[BM][NEXP];
    __shared__ int   sAny[NEXP];
    __shared__ int   sAidx[NEXP];
    __shared__ int   sAcnt;

    const int tid  = threadIdx.x;
    const int wave = tid >> 5;
    const int lane = tid & 31;
    const int half = lane >> 4;
    const int ln   = lane & 15;
    const int wm   = wave >> 2;          // 0..1 -> M offset wm*32
    const int wn   = wave & 3;           // 0..3 -> N offset wn*32
    const int t0   = blockIdx.x * BM;
    const int f0   = blockIdx.y * BN;
    const int r0   = wm * 32 + ln;       // A row held by this lane (subtile 0)
    const int nb0  = wn * 32;            // B col bases: nb0, nb0+16

    if (tid < NEXP) sAny[tid] = 0;
    __syncthreads();
#pragma unroll
    for (int i = 0; i < 2; ++i) {
        const int idx = tid + 256 * i;       // 512 = BM*NEXP entries
        const int m = idx >> 3, e = idx & 7;
        const float ce = combine[(size_t)(t0 + m) * NEXP + e];
        sComb[m][e] = ce;
        if (ce != 0.0f) atomicOr(&sAny[e], 1);
    }
    __syncthreads();
    if (tid == 0) {
        int n = 0;
        for (int e = 0; e < NEXP; ++e)
            if (sAny[e]) sAidx[n++] = e;
        sAcnt = n;
    }
    __syncthreads();
    const int Acnt = sAcnt;              // uniform; >= 1 always (top-2 gating)

    // ---- async stagers (per wave: X = 2 instr, W = 4 instr) ----
    auto issueX = [&](int k0, int xb) {
#pragma unroll
        for (int i = 0; i < 2; ++i) {
            const int j   = tid + 256 * i;
            const int row = j >> 3;
            const int c4  = (j & 7) * 4;
            cp16(&sX[xb][row][c4], x + (size_t)(t0 + row) * DDIM + k0 + c4);
        }
    };
    auto issueW = [&](int e, int k0, int wb) {
        const float* wsrc = ew + ((size_t)e * DDIM + k0) * FDIM + f0;
#pragma unroll
        for (int i = 0; i < 4; ++i) {
            const int j   = tid + 256 * i;
            const int row = j >> 5;
            const int c4  = (j & 31) * 4;
            cp16(&sW[wb][row][c4], wsrc + (size_t)row * FDIM + c4);
        }
    };

    v8f c00 = {}, c01 = {}, c10 = {}, c11 = {};

    // ---- pipeline prologue: tile p=0 ----
    issueX(0, 0);
    issueW(sAidx[0], 0, 0);

    const int nit = (DDIM / KC) * Acnt;
    int ei = 0, kblk = 0;

    for (int p = 0; p < nit; ++p) {
        // next-tile indices
        int ei_n = ei + 1, kblk_n = kblk;
        if (ei_n == Acnt) { ei_n = 0; kblk_n = kblk + 1; }

        // step 1: issue async loads for tile p+1 into the other buffer
        if (p + 1 < nit) {
            if (ei_n == 0) issueX(kblk_n * KC, kblk_n & 1);   // X first
            issueW(sAidx[ei_n], kblk_n * KC, (p + 1) & 1);    // then the 4 W's
            wait_async_le4();   // last 4 outstanding = tile p+1's W -> tile p complete
        } else {
            wait_async_le0();
        }
        __syncthreads();

        // step 2: WMMA on tile p
        {
            const int e  = sAidx[ei];
            const int xb = kblk & 1;
            const int wb = p & 1;
            const float (*Xb)[XP] = sX[xb];
            const float (*Wb)[WP] = sW[wb];
            const float ce0 = sComb[r0][e];
            const float ce1 = sComb[r0 + 16][e];

#pragma unroll
            for (int kk = 0; kk < KC; kk += 4) {
                const int kr = kk + 2 * half;
                v2f a0, a1, b0, b1;
                a0.x = Xb[r0][kr];        a0.y = Xb[r0][kr + 1];
                a1.x = Xb[r0 + 16][kr];   a1.y = Xb[r0 + 16][kr + 1];
                a0.x *= ce0; a0.y *= ce0;
                a1.x *= ce1; a1.y *= ce1;
                b0.x = Wb[kr][nb0 + ln];       b0.y = Wb[kr + 1][nb0 + ln];
                b1.x = Wb[kr][nb0 + 16 + ln];  b1.y = Wb[kr + 1][nb0 + 16 + ln];

                c00 = __builtin_amdgcn_wmma_f32_16x16x4_f32(false, a0, false, b0, (short)0, c00, false, false);
                c01 = __builtin_amdgcn_wmma_f32_16x16x4_f32(false, a0, false, b1, (short)0, c01, false, false);
                c10 = __builtin_amdgcn_wmma_f32_16x16x4_f32(false, a1, false, b0, (short)0, c10, false, false);
                c11 = __builtin_amdgcn_wmma_f32_16x16x4_f32(false, a1, false, b1, (short)0, c11, false, false);
            }
        }
        __syncthreads();

        ei = ei_n; kblk = kblk_n;
    }

    // combine-weighted bias + store (two column groups: fc0, fc0+16)
    const int fc0 = f0 + nb0 + ln;
    float bA[NEXP], bB[NEXP];
#pragma unroll
    for (int e = 0; e < NEXP; ++e) {
        bA[e] = eb[e * FDIM + fc0];
        bB[e] = eb[e * FDIM + fc0 + 16];
    }

#pragma unroll
    for (int r = 0; r < 8; ++r) {
        const int m1 = wm * 32 + half * 8 + r;        // M subtile 0
        const int m2 = m1 + 16;                        // M subtile 1
        float a00 = c00[r], a01 = c01[r];
        float a10 = c10[r], a11 = c11[r];
#pragma unroll
        for (int e = 0; e < NEXP; ++e) {
            a00 += sComb[m1][e] * bA[e];
            a01 += sComb[m1][e] * bB[e];
            a10 += sComb[m2][e] * bA[e];
            a11 += sComb[m2][e] * bB[e];
        }
        out[(size_t)(t0 + m1) * FDIM + fc0]      = a00;
        out[(size_t)(t0 + m1) * FDIM + fc0 + 16] = a01;
        out[(size_t)(t0 + m2) * FDIM + fc0]      = a10;
        out[(size_t)(t0 + m2) * FDIM + fc0 + 16] = a11;
    }
}

// ---------------------------------------------------------------------------
extern "C" void kernel_launch(void* const* d_in, const int* in_sizes, int n_in,
                              void* d_out, int out_size, void* d_ws, size_t ws_size,
                              hipStream_t stream)
{
    const float* x   = (const float*)d_in[0];
    const float* nx  = (const float*)d_in[1];
    const float* rw  = (const float*)d_in[2];
    const float* nrw = (const float*)d_in[3];
    const float* ew  = (const float*)d_in[4];
    const float* eb  = (const float*)d_in[5];
    float* out      = (float*)d_out;
    float* combine  = (float*)d_ws;     // [TOKENS][NEXP] floats = 128 KB

    // router: 8 waves/block -> 512 blocks cover 4096 tokens
    moe_router<<<dim3(TOKENS / 8), 256, 0, stream>>>(x, nx, rw, nrw, combine);

    // gated grouped GEMM: 64 token tiles x 8 F tiles
    dim3 grid(TOKENS / 64, FDIM / 128);
    moe_expert<<<grid, 256, 0, stream>>>(x, ew, eb, combine, out);
}